// ZoomAttention_26980984553616
// MI455X (gfx1250) — compile-verified
//
#include <hip/hip_runtime.h>
#include <math.h>

// Problem constants
constexpr int CB  = 8;      // batch
constexpr int CN  = 1024;   // seq len
constexpr int CD  = 768;    // model dim
constexpr int CH  = 12;     // heads
constexpr int CDH = 64;     // head dim
#define SCALE_Q 0.03608439182435161f   // 768^-0.5

typedef __attribute__((ext_vector_type(2))) float v2f;
typedef __attribute__((ext_vector_type(8))) float v8f;

// D = A(16x4) * B(4x16) + C(16x16), full fp32 via CDNA5 WMMA.
static __device__ __forceinline__ v8f wmma_f32(v2f a, v2f b, v8f c) {
    return __builtin_amdgcn_wmma_f32_16x16x4_f32(
        /*neg_a=*/false, a, /*neg_b=*/false, b,
        /*c_mod=*/(short)0, c, /*reuse_a=*/false, /*reuse_b=*/false);
}

// LDS row pads (multiples of 4 -> 16B-aligned b128 staging, 8B-aligned v2f frags,
// and conflict-free 16-lane fragment reads)
constexpr int PK = 36;   // pad for 32-wide K tiles (GEMM)
constexpr int PA = 68;   // pad for 64-wide tiles (attention)

// ---------------------------------------------------------------------------
// Kernel 1: QKV projection. C[8192,2304] = X[8192,768] @ W[768,2304] + bias,
// scattered into Q/K/V laid out as [B,H,N,DH].
// Block = 128 threads (4 waves). Block tile 64x64, wave tile 16x64.
// Double-buffered LDS pipeline; all WMMA operand reads are aligned b64.
// ---------------------------------------------------------------------------
__global__ __launch_bounds__(128) void qkv_gemm_kernel(
    const float* __restrict__ X, const float* __restrict__ W,
    const float* __restrict__ bias, float* __restrict__ qkv)
{
    __shared__ float Xs[2][64][PK];   // A tile: [m][k]
    __shared__ float Wt[2][64][PK];   // B tile TRANSPOSED: [n][k]

    const int ct = blockIdx.x;          // 0..35 column tile
    const int rt = blockIdx.y;          // 0..127 row tile
    const int tid  = threadIdx.x;
    const int wave = tid >> 5;
    const int lane = tid & 31;
    const int row0 = rt * 64, col0 = ct * 64;
    const int mrow  = lane & 15;
    const int khalf = (lane >> 4) << 1; // 0 or 2
    const int mhi   = (lane >> 4) << 3; // 0 or 8

    // staging coords
    const int xr_r = tid >> 3;          // 0..15 (+16j)
    const int xr_c = (tid & 7) << 2;    // 0..28
    const int w_n  = tid & 63;          // 0..63
    const int w_k0 = (tid >> 6) << 4;   // 0 or 16

    v8f vzero = {};
    v8f acc[4];
    #pragma unroll
    for (int s = 0; s < 4; ++s) acc[s] = vzero;

    float4 xreg[4];
    float  wreg[16];

    // ---- prologue: stage tile kt=0 into buffer 0
    #pragma unroll
    for (int j = 0; j < 4; ++j)
        xreg[j] = *(const float4*)(X + (size_t)(row0 + xr_r + 16 * j) * CD + xr_c);
    #pragma unroll
    for (int j = 0; j < 16; ++j)
        wreg[j] = W[(size_t)(w_k0 + j) * (3 * CD) + col0 + w_n];
    #pragma unroll
    for (int j = 0; j < 4; ++j)
        *(float4*)&Xs[0][xr_r + 16 * j][xr_c] = xreg[j];
    #pragma unroll
    for (int j = 0; j < 4; ++j) {
        float4 t = make_float4(wreg[4*j], wreg[4*j+1], wreg[4*j+2], wreg[4*j+3]);
        *(float4*)&Wt[0][w_n][w_k0 + 4 * j] = t;
    }
    __syncthreads();

    int buf = 0;
    for (int kt = 0; kt < CD; kt += 32) {
        const bool has_next = (kt + 32) < CD;
        if (has_next) {
            #pragma unroll
            for (int j = 0; j < 4; ++j)
                xreg[j] = *(const float4*)(X + (size_t)(row0 + xr_r + 16 * j) * CD + kt + 32 + xr_c);
            #pragma unroll
            for (int j = 0; j < 16; ++j)
                wreg[j] = W[(size_t)(kt + 32 + w_k0 + j) * (3 * CD) + col0 + w_n];
        }

        #pragma unroll
        for (int kk = 0; kk < 8; ++kk) {
            v2f a = *(const v2f*)&Xs[buf][wave * 16 + mrow][kk * 4 + khalf];
            #pragma unroll
            for (int s = 0; s < 4; ++s) {
                v2f b = *(const v2f*)&Wt[buf][s * 16 + mrow][kk * 4 + khalf];
                acc[s] = wmma_f32(a, b, acc[s]);
            }
        }

        if (has_next) {
            const int nb = buf ^ 1;
            #pragma unroll
            for (int j = 0; j < 4; ++j)
                *(float4*)&Xs[nb][xr_r + 16 * j][xr_c] = xreg[j];
            #pragma unroll
            for (int j = 0; j < 4; ++j) {
                float4 t = make_float4(wreg[4*j], wreg[4*j+1], wreg[4*j+2], wreg[4*j+3]);
                *(float4*)&Wt[nb][w_n][w_k0 + 4 * j] = t;
            }
        }
        __syncthreads();
        buf ^= 1;
    }

    // Scatter epilogue: column tile ct -> (three = ct/12, h = ct%12)
    const int three = ct / CH, h = ct % CH;
    const int b     = row0 >> 10;
    const int nbase = row0 & (CN - 1);
    float* dst = qkv + (size_t)three * ((size_t)CB * CH * CN * CDH);
    #pragma unroll
    for (int s = 0; s < 4; ++s) {
        const int dh = s * 16 + mrow;
        const float bv = bias[ct * 64 + dh];
        #pragma unroll
        for (int r = 0; r < 8; ++r) {
            const int m = wave * 16 + r + mhi;
            dst[((size_t)(b * CH + h) * CN + nbase + m) * CDH + dh] = acc[s][r] + bv;
        }
    }
}

// ---------------------------------------------------------------------------
// Kernel 2: flash attention per (b,h). 4 waves x 16 queries = 64-query tile.
// Streams 64-key chunks of K,V through LDS with online softmax.
// Ks kept [key][dh] (QK^T B-frag contiguous); V stored TRANSPOSED [dh][key]
// (PV B-frag contiguous). scores = (q.k)*SCALE - gamma[b]*dist[n,m].
// ---------------------------------------------------------------------------
__global__ __launch_bounds__(128) void attn_kernel(
    const float* __restrict__ q, const float* __restrict__ k,
    const float* __restrict__ v, const float* __restrict__ dist,
    const float* __restrict__ gamma, float* __restrict__ out)
{
    __shared__ float Ks [64][PA];     // [key][dh]
    __shared__ float Vst[64][PA];     // [dh][key]  (transposed V)
    __shared__ float Ps [4][16][PA];  // per-wave P tile [qrow][key]

    const int bh = blockIdx.y;
    const int b  = bh / CH, h = bh % CH;
    const int q0 = blockIdx.x * 64;
    const int tid  = threadIdx.x;
    const int wave = tid >> 5;
    const int lane = tid & 31;
    const int mrow  = lane & 15;
    const int khalf = (lane >> 4) << 1;
    const int mhi   = (lane >> 4) << 3;
    const float g = gamma[b];

    const float* qb = q + (size_t)bh * CN * CDH;
    const float* kb = k + (size_t)bh * CN * CDH;
    const float* vb = v + (size_t)bh * CN * CDH;

    // staging coords
    const int ks_key = tid >> 4;        // 0..7 (+8j)
    const int ks_c   = (tid & 15) << 2; // 0..60
    const int v_dh   = tid & 63;        // 0..63
    const int v_k0   = (tid >> 6) << 5; // 0 or 32

    // Preload this wave's Q rows as A-fragments (SCALE folded in), b64 loads
    v2f qa[16];
    #pragma unroll
    for (int kk = 0; kk < 16; ++kk) {
        v2f t = *(const v2f*)(qb + (size_t)(q0 + wave * 16 + mrow) * CDH + kk * 4 + khalf);
        qa[kk] = t * SCALE_Q;
    }

    v8f vzero = {};
    v8f oacc[4];
    #pragma unroll
    for (int s = 0; s < 4; ++s) oacc[s] = vzero;
    float rmax[8], lsum[8];
    #pragma unroll
    for (int r = 0; r < 8; ++r) { rmax[r] = -INFINITY; lsum[r] = 0.0f; }

    for (int kc = 0; kc < CN / 64; ++kc) {
        if (kc) __syncthreads();   // protect Ks/Vst from previous P@V readers

        // Stage K [key][dh] with float4 rows
        #pragma unroll
        for (int j = 0; j < 8; ++j) {
            const int key = ks_key + 8 * j;
            float4 kv = *(const float4*)(kb + (size_t)(kc * 64 + key) * CDH + ks_c);
            *(float4*)&Ks[key][ks_c] = kv;
        }
        // Stage V transposed: thread owns column dh, reads 32 keys (coalesced
        // across lanes), writes contiguous [dh][k0..k0+31]
        #pragma unroll
        for (int j = 0; j < 32; j += 4) {
            float a0 = vb[(size_t)(kc * 64 + v_k0 + j + 0) * CDH + v_dh];
            float a1 = vb[(size_t)(kc * 64 + v_k0 + j + 1) * CDH + v_dh];
            float a2 = vb[(size_t)(kc * 64 + v_k0 + j + 2) * CDH + v_dh];
            float a3 = vb[(size_t)(kc * 64 + v_k0 + j + 3) * CDH + v_dh];
            *(float4*)&Vst[v_dh][v_k0 + j] = make_float4(a0, a1, a2, a3);
        }
        __syncthreads();

        // S[16x64] = Q @ K^T  (B-frag = Ks[key][dh..dh+1], contiguous b64)
        v8f sacc[4];
        #pragma unroll
        for (int s = 0; s < 4; ++s) sacc[s] = vzero;
        #pragma unroll
        for (int kk = 0; kk < 16; ++kk) {
            v2f a = qa[kk];
            #pragma unroll
            for (int s = 0; s < 4; ++s) {
                v2f bf = *(const v2f*)&Ks[s * 16 + mrow][kk * 4 + khalf];
                sacc[s] = wmma_f32(a, bf, sacc[s]);
            }
        }

        // Bias + chunk row max
        float cmax[8];
        #pragma unroll
        for (int r = 0; r < 8; ++r) cmax[r] = -INFINITY;
        #pragma unroll
        for (int s = 0; s < 4; ++s) {
            const int keyg = kc * 64 + s * 16 + mrow;
            #pragma unroll
            for (int r = 0; r < 8; ++r) {
                const int qrow = q0 + wave * 16 + r + mhi;
                float sv = sacc[s][r] - g * dist[(size_t)qrow * CN + keyg];
                sacc[s][r] = sv;
                cmax[r] = fmaxf(cmax[r], sv);
            }
        }
        #pragma unroll
        for (int m = 1; m < 16; m <<= 1) {
            #pragma unroll
            for (int r = 0; r < 8; ++r)
                cmax[r] = fmaxf(cmax[r], __shfl_xor(cmax[r], m, 32));
        }

        // Online softmax rescale
        float nm[8], fac[8];
        #pragma unroll
        for (int r = 0; r < 8; ++r) {
            nm[r]  = fmaxf(rmax[r], cmax[r]);
            fac[r] = expf(rmax[r] - nm[r]);   // first chunk: exp(-inf)=0
            rmax[r] = nm[r];
            lsum[r] *= fac[r];
        }
        #pragma unroll
        for (int s = 0; s < 4; ++s)
            #pragma unroll
            for (int r = 0; r < 8; ++r) oacc[s][r] *= fac[r];

        // P = exp(S - m), row-sum, stage to per-wave LDS tile (A-layout bridge)
        float psum[8];
        #pragma unroll
        for (int r = 0; r < 8; ++r) psum[r] = 0.0f;
        #pragma unroll
        for (int s = 0; s < 4; ++s) {
            #pragma unroll
            for (int r = 0; r < 8; ++r) {
                float p = expf(sacc[s][r] - nm[r]);
                psum[r] += p;
                Ps[wave][r + mhi][s * 16 + mrow] = p;
            }
        }
        #pragma unroll
        for (int m = 1; m < 16; m <<= 1) {
            #pragma unroll
            for (int r = 0; r < 8; ++r) psum[r] += __shfl_xor(psum[r], m, 32);
        }
        #pragma unroll
        for (int r = 0; r < 8; ++r) lsum[r] += psum[r];

        __syncthreads();   // Ps visible across lanes

        // O += P[16x64] @ V[64x64]  (A = Ps contiguous, B = Vst contiguous)
        #pragma unroll
        for (int kk = 0; kk < 16; ++kk) {
            v2f a = *(const v2f*)&Ps[wave][mrow][kk * 4 + khalf];
            #pragma unroll
            for (int s = 0; s < 4; ++s) {
                v2f bf = *(const v2f*)&Vst[s * 16 + mrow][kk * 4 + khalf];
                oacc[s] = wmma_f32(a, bf, oacc[s]);
            }
        }
    }

    // Normalize and write attn output as [B,N,D] (head transpose folded in)
    #pragma unroll
    for (int r = 0; r < 8; ++r) lsum[r] = 1.0f / lsum[r];
    #pragma unroll
    for (int s = 0; s < 4; ++s) {
        const int dh = s * 16 + mrow;
        #pragma unroll
        for (int r = 0; r < 8; ++r) {
            const int n = q0 + wave * 16 + r + mhi;
            out[((size_t)(b * CN + n)) * CD + h * CDH + dh] = oacc[s][r] * lsum[r];
        }
    }
}

// ---------------------------------------------------------------------------
// Kernel 3: output projection. C[8192,768] = A[8192,768] @ Wp[768,768] + bp
// Same pipelined structure as kernel 1.
// ---------------------------------------------------------------------------
__global__ __launch_bounds__(128) void proj_gemm_kernel(
    const float* __restrict__ A, const float* __restrict__ W,
    const float* __restrict__ bias, float* __restrict__ C)
{
    __shared__ float Xs[2][64][PK];
    __shared__ float Wt[2][64][PK];

    const int ct = blockIdx.x;          // 0..11
    const int rt = blockIdx.y;          // 0..127
    const int tid  = threadIdx.x;
    const int wave = tid >> 5;
    const int lane = tid & 31;
    const int row0 = rt * 64, col0 = ct * 64;
    const int mrow  = lane & 15;
    const int khalf = (lane >> 4) << 1;
    const int mhi   = (lane >> 4) << 3;

    const int xr_r = tid >> 3;
    const int xr_c = (tid & 7) << 2;
    const int w_n  = tid & 63;
    const int w_k0 = (tid >> 6) << 4;

    v8f vzero = {};
    v8f acc[4];
    #pragma unroll
    for (int s = 0; s < 4; ++s) acc[s] = vzero;

    float4 xreg[4];
    float  wreg[16];

    #pragma unroll
    for (int j = 0; j < 4; ++j)
        xreg[j] = *(const float4*)(A + (size_t)(row0 + xr_r + 16 * j) * CD + xr_c);
    #pragma unroll
    for (int j = 0; j < 16; ++j)
        wreg[j] = W[(size_t)(w_k0 + j) * CD + col0 + w_n];
    #pragma unroll
    for (int j = 0; j < 4; ++j)
        *(float4*)&Xs[0][xr_r + 16 * j][xr_c] = xreg[j];
    #pragma unroll
    for (int j = 0; j < 4; ++j) {
        float4 t = make_float4(wreg[4*j], wreg[4*j+1], wreg[4*j+2], wreg[4*j+3]);
        *(float4*)&Wt[0][w_n][w_k0 + 4 * j] = t;
    }
    __syncthreads();

    int buf = 0;
    for (int kt = 0; kt < CD; kt += 32) {
        const bool has_next = (kt + 32) < CD;
        if (has_next) {
            #pragma unroll
            for (int j = 0; j < 4; ++j)
                xreg[j] = *(const float4*)(A + (size_t)(row0 + xr_r + 16 * j) * CD + kt + 32 + xr_c);
            #pragma unroll
            for (int j = 0; j < 16; ++j)
                wreg[j] = W[(size_t)(kt + 32 + w_k0 + j) * CD + col0 + w_n];
        }

        #pragma unroll
        for (int kk = 0; kk < 8; ++kk) {
            v2f a = *(const v2f*)&Xs[buf][wave * 16 + mrow][kk * 4 + khalf];
            #pragma unroll
            for (int s = 0; s < 4; ++s) {
                v2f b = *(const v2f*)&Wt[buf][s * 16 + mrow][kk * 4 + khalf];
                acc[s] = wmma_f32(a, b, acc[s]);
            }
        }

        if (has_next) {
            const int nb = buf ^ 1;
            #pragma unroll
            for (int j = 0; j < 4; ++j)
                *(float4*)&Xs[nb][xr_r + 16 * j][xr_c] = xreg[j];
            #pragma unroll
            for (int j = 0; j < 4; ++j) {
                float4 t = make_float4(wreg[4*j], wreg[4*j+1], wreg[4*j+2], wreg[4*j+3]);
                *(float4*)&Wt[nb][w_n][w_k0 + 4 * j] = t;
            }
        }
        __syncthreads();
        buf ^= 1;
    }

    #pragma unroll
    for (int s = 0; s < 4; ++s) {
        const int c = col0 + s * 16 + mrow;
        const float bv = bias[c];
        #pragma unroll
        for (int r = 0; r < 8; ++r) {
            const int m = wave * 16 + r + mhi;
            C[(size_t)(row0 + m) * CD + c] = acc[s][r] + bv;
        }
    }
}

// ---------------------------------------------------------------------------
extern "C" void kernel_launch(void* const* d_in, const int* in_sizes, int n_in,
                              void* d_out, int out_size, void* d_ws, size_t ws_size,
                              hipStream_t stream) {
    (void)in_sizes; (void)n_in; (void)out_size; (void)ws_size;
    const float* x      = (const float*)d_in[0];
    const float* gamma  = (const float*)d_in[1];
    const float* dist   = (const float*)d_in[2];
    const float* W_qkv  = (const float*)d_in[3];
    const float* b_qkv  = (const float*)d_in[4];
    const float* W_proj = (const float*)d_in[5];
    const float* b_proj = (const float*)d_in[6];
    float* out = (float*)d_out;

    const size_t QS = (size_t)CB * CH * CN * CDH;   // 6,291,456 floats per tensor
    float* qkv      = (float*)d_ws;                 // Q | K | V  (3*QS floats)
    float* attn_out = qkv + 3 * QS;                 // [B,N,D]    (QS floats)

    // 1) QKV projection -> Q/K/V in [B,H,N,DH]
    qkv_gemm_kernel<<<dim3(3 * CD / 64, CB * CN / 64), 128, 0, stream>>>(
        x, W_qkv, b_qkv, qkv);

    // 2) Flash attention with distance bias -> attn_out [B,N,D]
    attn_kernel<<<dim3(CN / 64, CB * CH), 128, 0, stream>>>(
        qkv, qkv + QS, qkv + 2 * QS, dist, gamma, attn_out);

    // 3) Output projection -> d_out
    proj_gemm_kernel<<<dim3(CD / 64, CB * CN / 64), 128, 0, stream>>>(
        attn_out, W_proj, b_proj, out);
}